// RF_eq_13572096655541
// MI455X (gfx1250) — compile-verified
//
#include <hip/hip_runtime.h>
#include <hip/hip_bf16.h>

// Elementwise PDE-residual map: out = f(x, y) with one exp + FMA chain.
// Memory-bound (AI ~2 flop/B, floor ~8.6us at 23.3 TB/s). Strategy: wave32
// streaming with b128 NT loads/stores, 4 points per thread, hardware
// v_exp_f32 via exp2. Matrix ops are inapplicable (no contraction dim).

// Native clang vector type: __builtin_nontemporal_* rejects HIP_vector_type.
typedef __attribute__((ext_vector_type(4))) float v4f;

static __device__ __forceinline__ float fast_exp(float t) {
    // exp(t) = exp2(t * log2(e)); v_exp_f32 computes 2^x in hardware.
    constexpr float LOG2E = 1.4426950408889634f;
#if __has_builtin(__builtin_amdgcn_exp2f)
    return __builtin_amdgcn_exp2f(t * LOG2E);
#else
    return __expf(t);
#endif
}

static __device__ __forceinline__ float point_eval(float x, float y) {
    const float x2 = x * x;
    const float y2 = y * y;
    // u = exp(-(x^2 + 2 y^2 + 1))
    const float u = fast_exp(-(x2 + 2.0f * y2 + 1.0f));
    // du2         = -4 u y
    // du11 + du22 = 4 u^2 x^2 + 16 u^2 y^2 - 6 u = u * (4 u (x^2 + 4 y^2) - 6)
    const float lap = u * fmaf(4.0f * u, fmaf(4.0f, y2, x2), -6.0f);
    // out = -((du11+du22)(1+2y^2) + 4 y * du2) + (1+x^2) u
    //     = -(lap * (1 + 2 y^2) - 16 u y^2) + (1 + x^2) u
    const float term = fmaf(lap, fmaf(2.0f, y2, 1.0f), -16.0f * u * y2);
    return fmaf(x2 + 1.0f, u, -term);
}

__global__ __launch_bounds__(256) void RF_eq_13572096655541_kernel(
    const v4f* __restrict__ in,   // interleaved (x,y) pairs: one v4f = 2 points
    v4f* __restrict__ out,        // 4 results per v4f
    int n4)                       // number of output float4s (N/4)
{
    const int i = blockIdx.x * blockDim.x + threadIdx.x;
    if (i >= n4) return;

    // Two b128 non-temporal loads -> 4 points; zero reuse, keep L2 clean.
    const v4f a = __builtin_nontemporal_load(&in[2 * i + 0]); // x0 y0 x1 y1
    const v4f b = __builtin_nontemporal_load(&in[2 * i + 1]); // x2 y2 x3 y3

    v4f r;
    r.x = point_eval(a.x, a.y);
    r.y = point_eval(a.z, a.w);
    r.z = point_eval(b.x, b.y);
    r.w = point_eval(b.z, b.w);

    __builtin_nontemporal_store(r, &out[i]);
}

extern "C" void kernel_launch(void* const* d_in, const int* in_sizes, int n_in,
                              void* d_out, int out_size, void* d_ws, size_t ws_size,
                              hipStream_t stream) {
    (void)n_in; (void)d_ws; (void)ws_size; (void)out_size;
    const v4f* in = (const v4f*)d_in[0];
    v4f* out = (v4f*)d_out;

    // in_sizes[0] is the flat element count of the (N,2) input -> N = count/2.
    const long long N = (long long)in_sizes[0] / 2;
    const int n4 = (int)(N / 4);            // N = 16,777,216 -> n4 = 4,194,304
    const int block = 256;                   // 8 wave32 waves per block
    const int grid = (n4 + block - 1) / block;

    RF_eq_13572096655541_kernel<<<grid, block, 0, stream>>>(in, out, n4);
}